// TimeVariantHyperLinearSSM_90314572300306
// MI455X (gfx1250) — compile-verified
//
#include <hip/hip_runtime.h>
#include <hip/hip_bf16.h>
#include <math.h>

// dims (compile-time, match reference)
constexpr int B_  = 64;
constexpr int T_  = 50;
constexpr int S_  = 256;
constexpr int A_  = 64;
constexpr int H_  = 256;

typedef float v2f __attribute__((ext_vector_type(2)));
typedef float v8f __attribute__((ext_vector_type(8)));

// D = A(16x4 f32) x B(4x16 f32) + C(16x16 f32), full-precision WMMA (CDNA5)
__device__ __forceinline__ v8f wmma4_f32(v2f a, v2f b, v8f c) {
  return __builtin_amdgcn_wmma_f32_16x16x4_f32(
      /*neg_a=*/false, a, /*neg_b=*/false, b,
      /*c_mod=*/(short)0, c, /*reuse_a=*/false, /*reuse_b=*/false);
}

// ---------------------------------------------------------------------------
// Kernel 0: hypernet activations for ALL (t,b,k) at once.
// HA[(t*B+b)*H+k] = tanh((context[b]+t/50)*W1a[k] + b1a[k]); HB likewise.
// ---------------------------------------------------------------------------
__global__ void ssm_hyper(const float* __restrict__ context,
                          const float* __restrict__ W1a, const float* __restrict__ b1a,
                          const float* __restrict__ W1b, const float* __restrict__ b1b,
                          float* __restrict__ HA, float* __restrict__ HB) {
  int idx = blockIdx.x * blockDim.x + threadIdx.x;
  if (idx >= T_ * B_ * H_) return;
  int k  = idx & (H_ - 1);
  int tb = idx >> 8;            // idx / H_
  int b  = tb % B_;
  int t  = tb / B_;
  float c = context[b] + (float)t * (1.0f / 50.0f);
  HA[idx] = tanhf(fmaf(c, W1a[k], b1a[k]));
  HB[idx] = tanhf(fmaf(c, W1b[k], b1b[k]));
}

// ---------------------------------------------------------------------------
// Kernel 1 (per step): split-k partials of
//   x_new[b,i] = sum_k ha[b,k] * (W2a[k,i,:] . x[b,:])
//             +  sum_k hb[b,k] * (W2b[k,i,:] . u[b,:])
//             +  (b2a[i,:].x[b,:]) + (b2b[i,:].u[b,:])            (k-chunk 7)
// Grid: 8 k-chunks x 16 i-tiles x 4 b-tiles = 512 blocks, 2 waves each.
// Each wave owns half of the j (K) contraction (the ha/hb scale distributes
// over the j-split), so waves are fully independent: no LDS, no barriers.
// WMMA tile: M=i (16), N=b (16), K=j streamed 4 at a time (f32 WMMA).
// ---------------------------------------------------------------------------
__global__ __launch_bounds__(64) void ssm_step_partial(
    const float* __restrict__ W2a, const float* __restrict__ b2a,
    const float* __restrict__ W2b, const float* __restrict__ b2b,
    const float* __restrict__ HA,  const float* __restrict__ HB,
    const float* __restrict__ xbase, int xstride,
    const float* __restrict__ ubase, int ustride,
    float* __restrict__ partials, int t) {
  const int lane = threadIdx.x & 31;
  const int wid  = threadIdx.x >> 5;      // 0/1: which half of the j-range
  const int blk  = blockIdx.x;            // kc*64 + it*4 + bt
  const int bt   = blk & 3;
  const int it   = (blk >> 2) & 15;
  const int kc   = blk >> 6;              // 0..7
  const int b0   = bt * 16;
  const int i0   = it * 16;
  const int MN   = lane & 15;             // row (A) / col (B) index for this lane
  const int half = lane >> 4;             // K sub-offset: lanes 0-15 -> K{0,1}, 16-31 -> K{2,3}
  const int jA0  = wid * 128;             // this wave's j-range for S-dim operands
  const int jB0  = wid * 32;              // this wave's j-range for A-dim operands

  // Preload B-fragments of x (16 batches x 128 per wave) and u (16 x 32).
  // B[Kj][N] layout mirrors the ISA f32 A layout: vgpr0/1 = K {2*half, 2*half+1}.
  const float* xrow = xbase + (size_t)(b0 + MN) * xstride + jA0 + half * 2;
  v2f xf[32];
#pragma unroll
  for (int f = 0; f < 32; ++f) {
    xf[f][0] = xrow[f * 4 + 0];
    xf[f][1] = xrow[f * 4 + 1];
  }
  const float* urow = ubase + (size_t)(b0 + MN) * ustride + jB0 + half * 2;
  v2f uf[8];
#pragma unroll
  for (int f = 0; f < 8; ++f) {
    uf[f][0] = urow[f * 4 + 0];
    uf[f][1] = urow[f * 4 + 1];
  }

  const float* haRow = HA + ((size_t)t * B_ + (b0 + MN)) * H_;
  const float* hbRow = HB + ((size_t)t * B_ + (b0 + MN)) * H_;

  v8f acc = {};
  const int kBeg = kc * 32, kEnd = kc * 32 + 32;
  for (int k = kBeg; k < kEnd; ++k) {
    // ---- A-part: T_k[i,b] += W2a[k, i0:i0+16, jA0:jA0+128] @ x^T
    const float* arow = W2a + (size_t)k * (S_ * S_) + (size_t)(i0 + MN) * S_ + jA0 + half * 2;
    v2f a0[16], a1[16];                   // two staged load groups
#pragma unroll
    for (int f = 0; f < 16; ++f) { a0[f][0] = arow[f * 4];      a0[f][1] = arow[f * 4 + 1]; }
#pragma unroll
    for (int f = 0; f < 16; ++f) { a1[f][0] = arow[64 + f * 4]; a1[f][1] = arow[64 + f * 4 + 1]; }
    v8f t0 = {}, t1 = {};                 // two chains to hide WMMA latency
#pragma unroll
    for (int f = 0; f < 16; f += 2) {
      t0 = wmma4_f32(a0[f],     xf[f],     t0);
      t1 = wmma4_f32(a0[f + 1], xf[f + 1], t1);
    }
#pragma unroll
    for (int f = 0; f < 16; f += 2) {
      t0 = wmma4_f32(a1[f],     xf[16 + f],     t0);
      t1 = wmma4_f32(a1[f + 1], xf[16 + f + 1], t1);
    }
    float sa = haRow[k];
#pragma unroll
    for (int r = 0; r < 8; ++r) acc[r] += sa * (t0[r] + t1[r]);

    // ---- B-part: W2b[k, i0:i0+16, jB0:jB0+32] @ u^T
    const float* brow = W2b + (size_t)k * (S_ * A_) + (size_t)(i0 + MN) * A_ + jB0 + half * 2;
    v2f bfr[8];
#pragma unroll
    for (int f = 0; f < 8; ++f) { bfr[f][0] = brow[f * 4]; bfr[f][1] = brow[f * 4 + 1]; }
    v8f s0 = {}, s1 = {};
#pragma unroll
    for (int f = 0; f < 8; f += 2) {
      s0 = wmma4_f32(bfr[f],     uf[f],     s0);
      s1 = wmma4_f32(bfr[f + 1], uf[f + 1], s1);
    }
    float sb = hbRow[k];
#pragma unroll
    for (int r = 0; r < 8; ++r) acc[r] += sb * (s0[r] + s1[r]);
  }

  if (kc == 7) {  // fold bias matvecs in as one extra "k" with weight 1
    const float* arow = b2a + (size_t)(i0 + MN) * S_ + jA0 + half * 2;
    v8f t0 = {}, t1 = {};
#pragma unroll
    for (int f = 0; f < 32; f += 2) {
      v2f c0; c0[0] = arow[f * 4];     c0[1] = arow[f * 4 + 1];
      v2f c1; c1[0] = arow[f * 4 + 4]; c1[1] = arow[f * 4 + 5];
      t0 = wmma4_f32(c0, xf[f],     t0);
      t1 = wmma4_f32(c1, xf[f + 1], t1);
    }
    const float* brow = b2b + (size_t)(i0 + MN) * A_ + jB0 + half * 2;
#pragma unroll
    for (int f = 0; f < 8; f += 2) {
      v2f c0; c0[0] = brow[f * 4];     c0[1] = brow[f * 4 + 1];
      v2f c1; c1[0] = brow[f * 4 + 4]; c1[1] = brow[f * 4 + 5];
      t0 = wmma4_f32(c0, uf[f],     t0);
      t1 = wmma4_f32(c1, uf[f + 1], t1);
    }
#pragma unroll
    for (int r = 0; r < 8; ++r) acc[r] += t0[r] + t1[r];
  }

  // partials[blk*2 + wid][lane][r]
  float* outp = partials + ((size_t)blk * 2 + wid) * 256 + lane * 8;
#pragma unroll
  for (int r = 0; r < 8; ++r) outp[r] = acc[r];
}

// ---------------------------------------------------------------------------
// Kernel 2 (per step): sum the 16 partial slices, write x_{t+1} into d_out.
// WMMA f32 D layout: vgpr r, lanes 0-15: (M=r,   N=lane),
//                            lanes 16-31:(M=r+8, N=lane-16).
// ---------------------------------------------------------------------------
__global__ void ssm_step_reduce(const float* __restrict__ partials,
                                float* __restrict__ out, int t) {
  int e = blockIdx.x * blockDim.x + threadIdx.x;   // 0 .. B*S-1
  if (e >= B_ * S_) return;
  int b = e >> 8;           // e / S_
  int i = e & (S_ - 1);
  int it = i >> 4, M = i & 15;
  int bt = b >> 4, N = b & 15;
  int lane = (M >= 8) ? (16 + N) : N;
  int r    = M & 7;
  float sum = 0.f;
#pragma unroll
  for (int kc = 0; kc < 8; ++kc) {
#pragma unroll
    for (int wid = 0; wid < 2; ++wid) {
      sum += partials[(((size_t)(kc * 64 + it * 4 + bt)) * 2 + wid) * 256 + lane * 8 + r];
    }
  }
  out[((size_t)b * T_ + t) * S_ + i] = sum;
}

// ---------------------------------------------------------------------------
extern "C" void kernel_launch(void* const* d_in, const int* in_sizes, int n_in,
                              void* d_out, int out_size, void* d_ws, size_t ws_size,
                              hipStream_t stream) {
  const float* x0      = (const float*)d_in[0];
  const float* context = (const float*)d_in[1];
  const float* us      = (const float*)d_in[2];
  const float* W1a     = (const float*)d_in[3];
  const float* b1a     = (const float*)d_in[4];
  const float* W2a     = (const float*)d_in[5];
  const float* b2a     = (const float*)d_in[6];
  const float* W1b     = (const float*)d_in[7];
  const float* b1b     = (const float*)d_in[8];
  const float* W2b     = (const float*)d_in[9];
  const float* b2b     = (const float*)d_in[10];
  float* out = (float*)d_out;

  // workspace layout (floats): HA | HB | partials (1024 slices x 256)
  float* HA       = (float*)d_ws;
  float* HB       = HA + (size_t)T_ * B_ * H_;
  float* partials = HB + (size_t)T_ * B_ * H_;

  int nHyp = T_ * B_ * H_;
  ssm_hyper<<<(nHyp + 255) / 256, 256, 0, stream>>>(context, W1a, b1a, W1b, b1b, HA, HB);

  for (int t = 0; t < T_; ++t) {
    // state input: x0 for t==0, else previous output row (strided through d_out)
    const float* xbase = (t == 0) ? x0 : (out + (size_t)(t - 1) * S_);
    int xstride        = (t == 0) ? S_ : T_ * S_;
    const float* ubase = us + (size_t)t * A_;    // row b: + b*(T_*A_)

    ssm_step_partial<<<512, 64, 0, stream>>>(W2a, b2a, W2b, b2b, HA, HB,
                                             xbase, xstride, ubase, T_ * A_,
                                             partials, t);
    ssm_step_reduce<<<(B_ * S_ + 255) / 256, 256, 0, stream>>>(partials, out, t);
  }
}